// RGBuvHistBlock_82403242541483
// MI455X (gfx1250) — compile-verified
//
#include <hip/hip_runtime.h>
#include <math.h>

// Problem constants (from reference)
#define NPIX 22500            // 150*150
#define HB   61
#define HB2  (HB * HB)        // 3721
#define SMAX 8                // max segments (K of the WMMA reduction = 8 = 2 x K4)

typedef float v2f __attribute__((ext_vector_type(2)));
typedef float v8f __attribute__((ext_vector_type(8)));

// Nearest-bin index with the reference's acceptance window.
// Bin centers: A[k] = -3.2 + k * (6.2951/60); accept if |v - A[k]| <= 3.2/61.
// Window (6.4/61) is slightly narrower than spacing (6.2951/60), so at most
// one bin matches -> nearest-center + window check is exact.
__device__ __forceinline__ int bin_index(float v) {
    const float A0      = -3.2f;
    const float STEP    = 6.2951f / 60.0f;
    const float INVSTEP = 60.0f / 6.2951f;
    const float HALFW   = 3.2f / 61.0f;     // EPS_BIN / 2
    float kr = rintf((v - A0) * INVSTEP);
    int k = (int)kr;
    if (k < 0 || k > 60) return -1;
    if (fabsf(v - (A0 + kr * STEP)) > HALFW) return -1;
    return k;
}

// Phase 1: each block = one (batch b, segment s). Builds all 3 pair-histograms
// for its pixel range in LDS (3 * 3721 * 4B = 44.6 KB) with ds_add_f32 atomics,
// then dumps partials to workspace: ws[((b*3+i)*S + s)*3721 + j].
__global__ void __launch_bounds__(256)
hist_partial_kernel(const float* __restrict__ X, float* __restrict__ ws, int S) {
    __shared__ float h[3 * HB2];
    const int b = blockIdx.x / S;
    const int s = blockIdx.x % S;

    for (int j = threadIdx.x; j < 3 * HB2; j += 256) h[j] = 0.0f;
    __syncthreads();

    const int per = (NPIX + S - 1) / S;
    const int n0  = s * per;
    const int n1  = min(NPIX, n0 + per);
    const float* Xb = X + (size_t)b * 3 * NPIX;
    const float EPSF = 2.2204e-16f;

    for (int n = n0 + (int)threadIdx.x; n < n1; n += 256) {
        float r  = Xb[n];
        float g  = Xb[NPIX + n];
        float bl = Xb[2 * NPIX + n];
        float iy = sqrtf(r * r + g * g + bl * bl);
        float l0 = logf(fabsf(r)  + EPSF);
        float l1 = logf(fabsf(g)  + EPSF);
        float l2 = logf(fabsf(bl) + EPSF);
        float d01 = l0 - l1, d02 = l0 - l2, d12 = l1 - l2;
        // pairs: (i=0,u=1,v=2) (i=1,u=0,v=2) (i=2,u=0,v=1)
        int k01 = bin_index(d01),  k02 = bin_index(d02);
        int k10 = bin_index(-d01), k12 = bin_index(d12);
        int k20 = bin_index(-d02), k21 = bin_index(-d12);
        if (k01 >= 0 && k02 >= 0) atomicAdd(&h[0 * HB2 + k01 * HB + k02], iy);
        if (k10 >= 0 && k12 >= 0) atomicAdd(&h[1 * HB2 + k10 * HB + k12], iy);
        if (k20 >= 0 && k21 >= 0) atomicAdd(&h[2 * HB2 + k20 * HB + k21], iy);
    }
    __syncthreads();

    for (int i = 0; i < 3; ++i) {
        float* dst = ws + ((size_t)(b * 3 + i) * S + s) * HB2;
        for (int j = threadIdx.x; j < HB2; j += 256)
            dst[j] = h[i * HB2 + j];
    }
}

// Phase 2: one block per (b,i); 8 waves each fold 16 histogram columns per
// chunk. Two chained V_WMMA_F32_16X16X4_F32 with an all-ones A sum the 8
// segment partials in full f32; row M=0 (lanes 0-15, acc VGPR 0) holds the
// exact column sums. Fused sqrt * C[i]/N epilogue.
//
// All loads are unconditional (addresses clamped in-bounds), so EXEC stays
// all-ones through the WMMA sequence; out-of-range contributions are zeroed
// with selects and the only predicated op is the final store.
__global__ void __launch_bounds__(256)
wmma_reduce_kernel(const float* __restrict__ ws, const float* __restrict__ C,
                   float* __restrict__ out, int S) {
    const int bi = blockIdx.x;            // b*3 + i, 0..191
    const int i  = bi % 3;
    const float scale = C[i] / (float)NPIX;
    const float* P = ws + (size_t)bi * (size_t)S * HB2;

    const int wave      = threadIdx.x >> 5;   // 0..7
    const int lane      = threadIdx.x & 31;
    const int half      = lane >> 4;          // lanes 0-15: K slots {0,1}; 16-31: {2,3}
    const int colInTile = lane & 15;

    // Per-lane segment rows for the two chained K=4 WMMAs:
    // first WMMA consumes segments {p0, p0+1}, second {p0+4, p0+5}.
    const int p0 = half * 2;                  // 0 or 2
    const bool v0 = (p0     < S), v1 = (p0 + 1 < S);
    const bool v2 = (p0 + 4 < S), v3 = (p0 + 5 < S);
    // Clamped row pointers: always within this bi's S*HB2 slice.
    const float* r0 = P + (size_t)min(p0,     S - 1) * HB2;
    const float* r1 = P + (size_t)min(p0 + 1, S - 1) * HB2;
    const float* r2 = P + (size_t)min(p0 + 4, S - 1) * HB2;
    const float* r3 = P + (size_t)min(p0 + 5, S - 1) * HB2;

    v2f ones = {1.0f, 1.0f};                  // A = 16x4 all-ones (layout-invariant)
    const int nchunks = (HB2 + 15) / 16;      // 233

    for (int chunk = wave; chunk < nchunks; chunk += 8) {   // wave-uniform trip count
        const int  col  = chunk * 16 + colInTile;
        const int  colc = min(col, HB2 - 1);                // clamped load address
        const bool ok   = col < HB2;

        float x0 = r0[colc], x1 = r1[colc];                 // unconditional loads
        float x2 = r2[colc], x3 = r3[colc];

        v2f bv0, bv1;
        bv0[0] = v0 ? x0 : 0.0f;  bv0[1] = v1 ? x1 : 0.0f;
        bv1[0] = v2 ? x2 : 0.0f;  bv1[1] = v3 ? x3 : 0.0f;

        v8f acc = {};
        // 8-arg pattern: (neg_a, A, neg_b, B, c_mod, C, reuse_a, reuse_b)
        acc = __builtin_amdgcn_wmma_f32_16x16x4_f32(false, ones, false, bv0,
                                                    (short)0, acc, false, false);
        acc = __builtin_amdgcn_wmma_f32_16x16x4_f32(false, ones, false, bv1,
                                                    (short)0, acc, false, false);

        if (half == 0 && ok) {
            out[(size_t)bi * HB2 + col] = sqrtf(fmaxf(acc[0], 0.0f)) * scale;
        }
    }
}

extern "C" void kernel_launch(void* const* d_in, const int* in_sizes, int n_in,
                              void* d_out, int out_size, void* d_ws, size_t ws_size,
                              hipStream_t stream) {
    (void)in_sizes; (void)n_in; (void)out_size;
    const float* X = (const float*)d_in[0];   // [64,3,150,150] f32
    const float* C = (const float*)d_in[1];   // [3] f32
    float* out = (float*)d_out;               // [64,3,61,61] f32
    float* ws  = (float*)d_ws;

    // Segments sized to available scratch: S * 192 * 3721 * 4 bytes needed.
    const size_t perSeg = (size_t)192 * HB2 * sizeof(float);  // ~2.86 MB
    int S = (ws_size >= perSeg) ? (int)(ws_size / perSeg) : 1;
    if (S > SMAX) S = SMAX;

    hist_partial_kernel<<<64 * S, 256, 0, stream>>>(X, ws, S);
    wmma_reduce_kernel<<<192, 256, 0, stream>>>(ws, C, out, S);
}